// HeteroLightGCNEns_59854664237640
// MI455X (gfx1250) — compile-verified
//
#include <hip/hip_runtime.h>

typedef __attribute__((ext_vector_type(2))) float v2f;
typedef __attribute__((ext_vector_type(8))) float v8f;

#define HDIM 64

// ---------------- degree count: deg[idx[e]] += 1 ----------------
__global__ void deg_kernel(const int* __restrict__ idx, float* __restrict__ deg, int E) {
  int i = blockIdx.x * blockDim.x + threadIdx.x;
  if (i < E) atomicAdd(&deg[idx[i]], 1.0f);
}

// ---------------- per-edge inverse norm: 1/sqrt(ds[si]*dt[ti]) ----------------
__global__ void norm_kernel(const int* __restrict__ si, const int* __restrict__ ti,
                            const float* __restrict__ ds, const float* __restrict__ dt,
                            float* __restrict__ inv, int E) {
  int e = blockIdx.x * blockDim.x + threadIdx.x;
  if (e < E) inv[e] = rsqrtf(ds[si[e]] * dt[ti[e]]);
}

// ---------------- edge scatter: acc[ti[e]] += x[si[e]] * inv[e] ----------------
// 16 threads per edge, float4 per thread (row = 64 f32 = 256B, L2-resident atomics)
__global__ void scatter_kernel(const float* __restrict__ x, const int* __restrict__ si,
                               const int* __restrict__ ti, const float* __restrict__ inv,
                               float* __restrict__ acc, int E) {
  long tid = (long)blockIdx.x * blockDim.x + threadIdx.x;
  long e = tid >> 4;
  if (e >= E) return;
  int c = ((int)tid & 15) << 2;
  int s = si[e], t = ti[e];
  float w = inv[e];
  const float4 v = *(const float4*)(x + (long)s * HDIM + c);
  float* dst = acc + (long)t * HDIM + c;
  atomicAdd(dst + 0, v.x * w);
  atomicAdd(dst + 1, v.y * w);
  atomicAdd(dst + 2, v.z * w);
  atomicAdd(dst + 3, v.w * w);
}

// ---------------- x = relu(x * scale) ----------------
__global__ void relu_scale_kernel(float* __restrict__ x, float scale, long n) {
  long i = (long)blockIdx.x * blockDim.x + threadIdx.x;
  if (i < n) x[i] = fmaxf(x[i] * scale, 0.0f);
}

// ---------------- out init with biases ----------------
__global__ void init_out_kernel(float* __restrict__ out, const float* __restrict__ bu,
                                const float* __restrict__ bs, int NU, int NS) {
  int i = blockIdx.x * blockDim.x + threadIdx.x;
  if (i < NU) out[i] = bu[0];
  else if (i < NU + NS) out[i] = bs[0];
}

// ---------------- projection via V_WMMA_F32_16X16X4_F32 ----------------
// out[n] += scale * dot(x[n][0:64], w[0:64]) for 16 nodes per wave.
// A (16x4 f32): lanes 0-15 hold K=0,1; lanes 16-31 hold K=2,3 (ISA 7.12.2).
// B (4x16 f32): lane n<16 holds B[0][n],B[1][n]; lane 16+n holds B[2][n],B[3][n].
// Weights go in column 0 of B -> D[:,0] in lanes 0 (M=0..7) and 16 (M=8..15).
__global__ void proj_wmma_kernel(const float* __restrict__ x, const float* __restrict__ w,
                                 float* __restrict__ out, int N, float scale) {
  int lane = threadIdx.x;          // wave32
  int row = lane & 15;
  int khalf = (lane >> 4) << 1;    // 0 for lanes 0-15, 2 for lanes 16-31
  long base = (long)blockIdx.x * 16;
  long r = base + row;
  if (r >= N) r = N - 1;           // clamp reads; EXEC stays all-1 for WMMA
  const float* xr = x + r * HDIM + khalf;
  float wm = (row == 0) ? scale : 0.0f;  // select (no branch) keeps EXEC full
  v8f c = {};
#pragma unroll
  for (int s = 0; s < 16; ++s) {
    v2f a;
    a.x = xr[4 * s + 0];
    a.y = xr[4 * s + 1];
    v2f b;
    b.x = w[4 * s + khalf + 0] * wm;
    b.y = w[4 * s + khalf + 1] * wm;
    c = __builtin_amdgcn_wmma_f32_16x16x4_f32(false, a, false, b, (short)0, c,
                                              false, false);
  }
  if (row == 0) {
    long mbase = base + (long)((lane >> 4) << 3);  // lane0 -> M=0..7, lane16 -> M=8..15
#pragma unroll
    for (int i = 0; i < 8; ++i)
      if (mbase + i < N) out[mbase + i] += c[i];
  }
}

static inline unsigned cdiv(long a, long b) { return (unsigned)((a + b - 1) / b); }

extern "C" void kernel_launch(void* const* d_in, const int* in_sizes, int n_in,
                              void* d_out, int out_size, void* d_ws, size_t ws_size,
                              hipStream_t stream) {
  (void)n_in; (void)out_size; (void)ws_size;

  const float* w_user = (const float*)d_in[14];
  const float* b_user = (const float*)d_in[15];
  const float* w_spot = (const float*)d_in[16];
  const float* b_spot = (const float*)d_in[17];
  const int* us_src = (const int*)d_in[18]; const int* us_dst = (const int*)d_in[19];
  const int* sc_src = (const int*)d_in[20]; const int* sc_dst = (const int*)d_in[21];
  const int* sci_src = (const int*)d_in[22]; const int* sci_dst = (const int*)d_in[23];
  const int* cp_src = (const int*)d_in[24]; const int* cp_dst = (const int*)d_in[25];

  const int NU = in_sizes[0] / HDIM, NS = in_sizes[1] / HDIM, NC = in_sizes[4] / HDIM,
            NCI = in_sizes[7] / HDIM, NP = in_sizes[8] / HDIM;
  const int Es[4] = {in_sizes[18], in_sizes[20], in_sizes[22], in_sizes[24]};

  // ---- workspace bump allocator (256B aligned) ----
  char* p = (char*)d_ws;
  auto alloc = [&](size_t nf) -> float* {
    float* r = (float*)p;
    p += ((nf * sizeof(float) + 255) & ~(size_t)255);
    return r;
  };

  // ---- degrees + per-edge inverse norms (computed once, shared by branches) ----
  const int* esrc[4] = {us_src, sc_src, sci_src, cp_src};
  const int* edst[4] = {us_dst, sc_dst, sci_dst, cp_dst};
  const int NsrcT[4] = {NU, NS, NS, NCI};
  const int NdstT[4] = {NS, NC, NCI, NP};
  float *degA[4], *degB[4], *invN[4];
  for (int k = 0; k < 4; ++k) {
    degA[k] = alloc(NsrcT[k]);
    degB[k] = alloc(NdstT[k]);
    invN[k] = alloc(Es[k]);
  }
  for (int k = 0; k < 4; ++k) {
    hipMemsetAsync(degA[k], 0, (size_t)NsrcT[k] * sizeof(float), stream);
    hipMemsetAsync(degB[k], 0, (size_t)NdstT[k] * sizeof(float), stream);
  }
  for (int k = 0; k < 4; ++k) {
    deg_kernel<<<cdiv(Es[k], 256), 256, 0, stream>>>(esrc[k], degA[k], Es[k]);
    deg_kernel<<<cdiv(Es[k], 256), 256, 0, stream>>>(edst[k], degB[k], Es[k]);
  }
  for (int k = 0; k < 4; ++k)
    norm_kernel<<<cdiv(Es[k], 256), 256, 0, stream>>>(esrc[k], edst[k], degA[k], degB[k],
                                                      invN[k], Es[k]);

  // ---- node buffers (ping-pong per type: 0=user 1=spot 2=cat 3=city 4=pref) ----
  const int Nt[5] = {NU, NS, NC, NCI, NP};
  float *bufA[5], *bufB[5];
  for (int t = 0; t < 5; ++t) {
    bufA[t] = alloc((size_t)Nt[t] * HDIM);
    bufB[t] = alloc((size_t)Nt[t] * HDIM);
  }

  // ---- relation table: {src type, tgt type, edge-type k, si, ti} ----
  struct Rel { int s, t, k; const int* si; const int* ti; };
  const Rel R[8] = {
      {0, 1, 0, us_src, us_dst},   // us_f  user->spot
      {1, 0, 0, us_dst, us_src},   // us_r  spot->user
      {1, 2, 1, sc_src, sc_dst},   // sc_f  spot->cat
      {2, 1, 1, sc_dst, sc_src},   // sc_r  cat->spot
      {1, 3, 2, sci_src, sci_dst}, // sci_f spot->city
      {3, 1, 2, sci_dst, sci_src}, // sci_r city->spot
      {3, 4, 3, cp_src, cp_dst},   // cp_f  city->pref
      {4, 3, 3, cp_dst, cp_src},   // cp_r  pref->city
  };
  const int brel[4][8] = {{0, 1, 0, 0, 0, 0, 0, 0},
                          {0, 1, 2, 3, 0, 0, 0, 0},
                          {0, 1, 4, 5, 6, 7, 0, 0},
                          {0, 1, 2, 3, 4, 5, 6, 7}};
  const int bn[4] = {2, 4, 6, 8};
  const float divs[4][5] = {{1, 1, 0, 0, 0}, {1, 2, 1, 0, 0}, {1, 2, 0, 2, 1}, {1, 3, 1, 2, 1}};
  const int pres[4][5] = {{1, 1, 0, 0, 0}, {1, 1, 1, 0, 0}, {1, 1, 0, 1, 1}, {1, 1, 1, 1, 1}};
  const float* x0[4][5] = {
      {(const float*)d_in[0], (const float*)d_in[1], nullptr, nullptr, nullptr},
      {(const float*)d_in[2], (const float*)d_in[3], (const float*)d_in[4], nullptr, nullptr},
      {(const float*)d_in[5], (const float*)d_in[6], nullptr, (const float*)d_in[7],
       (const float*)d_in[8]},
      {(const float*)d_in[9], (const float*)d_in[10], (const float*)d_in[13],
       (const float*)d_in[11], (const float*)d_in[12]},
  };

  // ---- output: init with biases; branches accumulate dot-products into it ----
  float* out = (float*)d_out;
  init_out_kernel<<<cdiv((long)NU + NS, 256), 256, 0, stream>>>(out, b_user, b_spot, NU, NS);

  const float inv3 = 1.0f / 3.0f;  // layer-mean folded into projection

  for (int b = 0; b < 4; ++b) {
    // projection of the initial embeddings (x0 term of the layer mean)
    proj_wmma_kernel<<<cdiv(NU, 16), 32, 0, stream>>>(x0[b][0], w_user + 64 * b, out, NU, inv3);
    proj_wmma_kernel<<<cdiv(NS, 16), 32, 0, stream>>>(x0[b][1], w_spot + 64 * b, out + NU, NS, inv3);

    const float* src[5];
    for (int t = 0; t < 5; ++t) src[t] = x0[b][t];

    for (int layer = 0; layer < 2; ++layer) {
      float** tgt = layer ? bufB : bufA;
      for (int t = 0; t < 5; ++t)
        if (pres[b][t])
          hipMemsetAsync(tgt[t], 0, (size_t)Nt[t] * HDIM * sizeof(float), stream);
      for (int i = 0; i < bn[b]; ++i) {
        const Rel& r = R[brel[b][i]];
        long nthreads = (long)Es[r.k] * 16;
        scatter_kernel<<<cdiv(nthreads, 256), 256, 0, stream>>>(src[r.s], r.si, r.ti,
                                                                invN[r.k], tgt[r.t], Es[r.k]);
      }
      for (int t = 0; t < 5; ++t)
        if (pres[b][t]) {
          long n = (long)Nt[t] * HDIM;
          relu_scale_kernel<<<cdiv(n, 256), 256, 0, stream>>>(tgt[t], 1.0f / divs[b][t], n);
        }
      // projection of this layer's user/spot outputs (l1 / l2 terms of the mean)
      proj_wmma_kernel<<<cdiv(NU, 16), 32, 0, stream>>>(tgt[0], w_user + 64 * b, out, NU, inv3);
      proj_wmma_kernel<<<cdiv(NS, 16), 32, 0, stream>>>(tgt[1], w_spot + 64 * b, out + NU, NS, inv3);
      for (int t = 0; t < 5; ++t)
        if (pres[b][t]) src[t] = tgt[t];
    }
  }
}